// Raymarcher_6536940224696
// MI455X (gfx1250) — compile-verified
//
#include <hip/hip_runtime.h>
#include <math.h>

// Problem constants (match reference)
#define BATCH   2
#define HH      256
#define WW      256
#define DD      128
#define NSTEPS  64

static constexpr float VOLRADIUS  = 256.0f;
static constexpr float DT         = 14.0f / 256.0f;   // DT_RAW / VOLRADIUS
static constexpr float STEPJITTER = 0.01f;

#if __has_builtin(__builtin_amdgcn_tensor_load_to_lds) && \
    __has_builtin(__builtin_amdgcn_s_wait_tensorcnt)
#define HAVE_TDM 1
#else
#define HAVE_TDM 0
#endif

typedef unsigned int v4u __attribute__((ext_vector_type(4)));
typedef int          v8i __attribute__((ext_vector_type(8)));
typedef int          v4i __attribute__((ext_vector_type(4)));

// ---------------------------------------------------------------------------
// Pass 1: relayout template [B,C,D,D,D] (8 MB channel stride) into AoS
//         [B,D,D,D,C] so each trilerp corner becomes ONE b128 load.
// Plain-load variant (fallback).
// ---------------------------------------------------------------------------
__global__ void __launch_bounds__(256)
relayout_kernel(const float* __restrict__ tmpl, float4* __restrict__ volI) {
    const int id = blockIdx.x * blockDim.x + threadIdx.x;   // b,z,y,x flat
    const int cs = DD * DD * DD;                            // channel stride
    const int x  =  id        & (DD - 1);
    const int y  = (id >> 7)  & (DD - 1);
    const int z  = (id >> 14) & (DD - 1);
    const int b  =  id >> 21;
    const size_t src = (size_t)b * 4 * cs + (size_t)((z * DD + y) * DD + x);
    volI[id] = make_float4(tmpl[src],
                           tmpl[src + cs],
                           tmpl[src + 2 * (size_t)cs],
                           tmpl[src + 3 * (size_t)cs]);
}

#if HAVE_TDM
// ---------------------------------------------------------------------------
// TDM variant: one tensor_load_to_lds per block pulls a 2D tile
//   tile_dim0 = 2048 contiguous floats (16 y-rows x 128 x)
//   tile_dim1 = 4 lines (channels), line stride = D^3 elements (8 MB)
// into LDS (4 x 8 KB), then 256 threads emit interleaved float4s.
// This toolchain (clang-23 / therock) exposes the 6-arg builtin:
//   (uint32x4, int32x8, int32x4, int32x4, int32x8, i32 cpol)
// ---------------------------------------------------------------------------
#define TDM_ROWS 16
__global__ void __launch_bounds__(256)
relayout_tdm_kernel(const float* __restrict__ tmpl, float4* __restrict__ volI) {
    constexpr int NELEM = DD * TDM_ROWS;          // 2048 floats per channel
    __shared__ float lds[4 * NELEM];              // 32 KB

    const int blk   = blockIdx.x;                 // ((b*DD)+z)*(DD/16)+ytile
    const int ytile =  blk        & (DD / TDM_ROWS - 1);
    const int z     = (blk >> 3)  & (DD - 1);
    const int b     =  blk >> 10;
    const int y0    = ytile * TDM_ROWS;

    if (threadIdx.x < 32) {                       // wave 0 issues the DMA
        // LDS byte offset = low 32 bits of the flat address (aperture is [63:32])
        const unsigned lds_base = (unsigned)(uintptr_t)(&lds[0]);
        const unsigned long long ga =
            (unsigned long long)(uintptr_t)tmpl +
            4ull * ((unsigned long long)b * 4 * DD * DD * DD +
                    (unsigned long long)(z * DD + y0) * DD);

        v4u g0;                                   // D# group 0 (128 b)
        g0[0] = 1u;                               // count=1, user mode
        g0[1] = lds_base;                         // lds_addr (bytes)
        g0[2] = (unsigned)(ga & 0xffffffffu);     // global_addr[31:0]
        g0[3] = (unsigned)((ga >> 32) & 0x01ffffffu) | 0x80000000u; // [56:32] | type=2

        v8i g1;                                   // D# group 1 (256 b)
        g1[0] = 0x00020000;                       // wg_mask=0, data_size=2 (4 B)
        g1[1] = (int)((unsigned)NELEM << 16);     // tensor_dim0[15:0] in [31:16]
        g1[2] = (int)(((unsigned)NELEM >> 16) | (4u << 16));  // dim0 hi | tensor_dim1=4
        g1[3] = (int)((unsigned)NELEM << 16);     // tile_dim0 = 2048
        g1[4] = 4;                                // tile_dim1=4 (channels), tile_dim2=0
        g1[5] = DD * DD * DD;                     // tensor_dim0_stride = channel stride
        g1[6] = 0;                                // stride0 hi, stride1 lo
        g1[7] = 0;                                // stride1 hi

        v4i g2;                                   // D# group 2: benign (2D tile)
        g2[0] = 1; g2[1] = 1; g2[2] = 0; g2[3] = 0;
        v4i g3;                                   // D# group 3: benign
        g3[0] = 0; g3[1] = 1 << 16; g3[2] = 0; g3[3] = 0;

        v8i g4 = {0, 0, 0, 0, 0, 0, 0, 0};        // extra operand (zero-filled,
                                                  // probe-verified benign form)

        __builtin_amdgcn_tensor_load_to_lds(g0, g1, g2, g3, g4, 0);
        __builtin_amdgcn_s_wait_tensorcnt(0);
    }
    __syncthreads();

    float4* outp = volI + ((size_t)(b * DD + z) * DD + y0) * DD;
    #pragma unroll
    for (int i = threadIdx.x; i < NELEM; i += 256) {
        outp[i] = make_float4(lds[i],
                              lds[NELEM + i],
                              lds[2 * NELEM + i],
                              lds[3 * NELEM + i]);
    }
}
#endif // HAVE_TDM

__device__ __forceinline__ float4 lerp4(float4 a, float4 b, float f) {
    return make_float4(a.x + f * (b.x - a.x),
                       a.y + f * (b.y - a.y),
                       a.z + f * (b.z - a.z),
                       a.w + f * (b.w - a.w));
}

__device__ __forceinline__ float4 fetchS(const float* __restrict__ v, int idx) {
    const int cs = DD * DD * DD;
    return make_float4(v[idx], v[idx + cs], v[idx + 2 * cs], v[idx + 3 * cs]);
}

// ---------------------------------------------------------------------------
// Pass 2: one thread per ray, 64-step march. Gather-latency bound; the 64 MB
// volume is L2-resident (192 MB L2), so we rely on wave parallelism +
// unroll-2 pipelining + global_prefetch_b8 of the next step's corner.
// ---------------------------------------------------------------------------
template <bool ILV>
__global__ void __launch_bounds__(256)
march_kernel(const float*  __restrict__ raydir,
             const float*  __restrict__ viewpos,
             const float4* __restrict__ volI,
             const float*  __restrict__ volS,
             const float*  __restrict__ tmin_noise,
             const float*  __restrict__ step_noise,
             float*        __restrict__ out)
{
    const int npix = HH * WW;
    const int nray = BATCH * npix;
    const int ray  = blockIdx.x * blockDim.x + threadIdx.x;
    if (ray >= nray) return;
    const int b   = ray / npix;
    const int pix = ray - b * npix;

    // ray direction + normalized view position
    const float* rdp = raydir + (size_t)ray * 3;
    const float rdx = rdp[0], rdy = rdp[1], rdz = rdp[2];
    const float inv = 1.0f / VOLRADIUS;
    const float vpx = viewpos[b * 3 + 0] * inv;
    const float vpy = viewpos[b * 3 + 1] * inv;
    const float vpz = viewpos[b * 3 + 2] * inv;

    // slab intersection with [-1,1]^3
    const float ax1 = (-1.0f - vpx) / rdx, ax2 = (1.0f - vpx) / rdx;
    const float ay1 = (-1.0f - vpy) / rdy, ay2 = (1.0f - vpy) / rdy;
    const float az1 = (-1.0f - vpz) / rdz, az2 = (1.0f - vpz) / rdz;
    float tmin = fmaxf(fmaxf(fminf(ax1, ax2), fminf(ay1, ay2)), fminf(az1, az2));
    float tmax = fminf(fminf(fmaxf(ax1, ax2), fmaxf(ay1, ay2)), fmaxf(az1, az2));
    const bool inter = tmin < tmax;
    tmin = inter ? tmin : 0.0f;
    tmax = inter ? tmax : 0.0f;
    tmin = tmin - DT * tmin_noise[ray];

    float px = vpx + rdx * tmin;
    float py = vpy + rdy * tmin;
    float pz = vpz + rdz * tmin;
    float t  = tmin;

    float accr = 0.0f, accg = 0.0f, accb = 0.0f, alpha = 0.0f;

    const float4* volb  = ILV ? (volI + (size_t)b * DD * DD * DD) : nullptr;
    const float*  volsb = volS + (size_t)b * 4 * DD * DD * DD;
    const float   scale = 0.5f * (float)(DD - 1);

    #pragma unroll 2
    for (int s = 0; s < NSTEPS; ++s) {
        const bool valid = (px > -1.0f) && (px < 1.0f) &&
                           (py > -1.0f) && (py < 1.0f) &&
                           (pz > -1.0f) && (pz < 1.0f);
        const bool alive = t < tmax;
        const float vf = (valid && alive) ? 1.0f : 0.0f;

        // trilerp coordinates (clamped -> gathers always in-bounds)
        const float gx = (px + 1.0f) * scale;
        const float gy = (py + 1.0f) * scale;
        const float gz = (pz + 1.0f) * scale;
        const float g0x = fminf(fmaxf(floorf(gx), 0.0f), (float)(DD - 2));
        const float g0y = fminf(fmaxf(floorf(gy), 0.0f), (float)(DD - 2));
        const float g0z = fminf(fmaxf(floorf(gz), 0.0f), (float)(DD - 2));
        const float fx = fminf(fmaxf(gx - g0x, 0.0f), 1.0f);
        const float fy = fminf(fmaxf(gy - g0y, 0.0f), 1.0f);
        const float fz = fminf(fmaxf(gz - g0z, 0.0f), 1.0f);
        const int x0 = (int)g0x, y0 = (int)g0y, z0 = (int)g0z;

        const int base = (z0 * DD + y0) * DD + x0;   // +1:x, +DD:y, +DD*DD:z
        float4 c000, c001, c010, c011, c100, c101, c110, c111;
        if (ILV) {
            c000 = volb[base];
            c001 = volb[base + 1];
            c010 = volb[base + DD];
            c011 = volb[base + DD + 1];
            c100 = volb[base + DD * DD];
            c101 = volb[base + DD * DD + 1];
            c110 = volb[base + DD * DD + DD];
            c111 = volb[base + DD * DD + DD + 1];
        } else {
            c000 = fetchS(volsb, base);
            c001 = fetchS(volsb, base + 1);
            c010 = fetchS(volsb, base + DD);
            c011 = fetchS(volsb, base + DD + 1);
            c100 = fetchS(volsb, base + DD * DD);
            c101 = fetchS(volsb, base + DD * DD + 1);
            c110 = fetchS(volsb, base + DD * DD + DD);
            c111 = fetchS(volsb, base + DD * DD + DD + 1);
        }

        // step size depends only on noise -> trajectory is sample-independent
        const float noise = step_noise[(size_t)s * nray + ray];
        const float step  = DT * __expf(STEPJITTER * noise);

        // blend tree
        const float4 cx00 = lerp4(c000, c001, fx);
        const float4 cx01 = lerp4(c010, c011, fx);
        const float4 cx10 = lerp4(c100, c101, fx);
        const float4 cx11 = lerp4(c110, c111, fx);
        const float4 cy0  = lerp4(cx00, cx01, fy);
        const float4 cy1  = lerp4(cx10, cx11, fy);
        const float4 samp = lerp4(cy0,  cy1,  fz);

        const float salpha  = fmaxf(samp.w, 0.0f);
        const float contrib = (fminf(alpha + salpha * step, 1.0f) - alpha) * vf;
        accr  += samp.x * contrib;
        accg  += samp.y * contrib;
        accb  += samp.z * contrib;
        alpha += contrib;

        // advance and prefetch next step's base corner (gfx1250 global_prefetch_b8)
        px += rdx * step;  py += rdy * step;  pz += rdz * step;
        t  += step;
        if (ILV) {
            const float nx = fminf(fmaxf(floorf((px + 1.0f) * scale), 0.0f), (float)(DD - 2));
            const float ny = fminf(fmaxf(floorf((py + 1.0f) * scale), 0.0f), (float)(DD - 2));
            const float nz = fminf(fmaxf(floorf((pz + 1.0f) * scale), 0.0f), (float)(DD - 2));
            const int nbase = (((int)nz * DD + (int)ny) * DD + (int)nx);
            __builtin_prefetch((const void*)(volb + nbase), 0, 3);
        }
    }

    // outputs: rgb [B,3,H,W] then alpha [B,1,H,W]
    out[((size_t)(b * 3 + 0)) * npix + pix] = accr;
    out[((size_t)(b * 3 + 1)) * npix + pix] = accg;
    out[((size_t)(b * 3 + 2)) * npix + pix] = accb;
    out[(size_t)BATCH * 3 * npix + (size_t)b * npix + pix] = alpha;
}

extern "C" void kernel_launch(void* const* d_in, const int* in_sizes, int n_in,
                              void* d_out, int out_size, void* d_ws, size_t ws_size,
                              hipStream_t stream)
{
    const float* raydir  = (const float*)d_in[0];   // (B,H,W,3)
    const float* viewpos = (const float*)d_in[1];   // (B,3)
    const float* tmpl    = (const float*)d_in[2];   // (B,C,D,D,D)
    const float* tminn   = (const float*)d_in[3];   // (B,H,W)
    const float* stepn   = (const float*)d_in[4];   // (NSTEPS,B,H,W)
    float* out = (float*)d_out;

    const int nray = BATCH * HH * WW;                       // 131072
    const int nvox = BATCH * DD * DD * DD;                  // 4194304
    const size_t need = (size_t)nvox * sizeof(float4);      // 64 MB

    if (ws_size >= need) {
        float4* volI = (float4*)d_ws;
#if HAVE_TDM
        const int ntile = BATCH * DD * (DD / TDM_ROWS);     // 2048 blocks
        relayout_tdm_kernel<<<ntile, 256, 0, stream>>>(tmpl, volI);
#else
        relayout_kernel<<<nvox / 256, 256, 0, stream>>>(tmpl, volI);
#endif
        march_kernel<true><<<nray / 256, 256, 0, stream>>>(
            raydir, viewpos, volI, tmpl, tminn, stepn, out);
    } else {
        // workspace too small: march directly on the strided layout
        march_kernel<false><<<nray / 256, 256, 0, stream>>>(
            raydir, viewpos, nullptr, tmpl, tminn, stepn, out);
    }
}